// GCN_81312320848287
// MI455X (gfx1250) — compile-verified
//
#include <hip/hip_runtime.h>
#include <hip/hip_bf16.h>

typedef __attribute__((ext_vector_type(2))) float v2f;
typedef __attribute__((ext_vector_type(8))) float v8f;

#define N_NODES 50000
#define N_EDGES 800000
#define N_GRAPH 512
#define F_IN    128
#define HID     64
#define EDIM    7
#define WSTRIDE 72   // LDS row stride (HID + 8) -> conflict-free WMMA B reads

#if defined(__has_builtin)
#if __has_builtin(__builtin_amdgcn_global_load_async_to_lds_b32)
#define HAVE_ASYNC_LDS 1
#endif
#if __has_builtin(__builtin_amdgcn_s_wait_asynccnt)
#define HAVE_WAIT_ASYNC 1
#endif
#endif

// Builtin expects addrspace(1) int* source and addrspace(3) int* destination.
#define AS_GLOBAL_I(p) ((__attribute__((address_space(1))) int*)(p))
#define AS_LDS_I(p)    ((__attribute__((address_space(3))) int*)(p))

// ---------------------------------------------------------------------------
// xl = x @ Wl , xr = x @ Wr  via V_WMMA_F32_16X16X4_F32
// Weights staged once per block into LDS (async-to-LDS when available),
// B operands then come from ds_load. One wave per 16-row tile; blockIdx.y
// selects Wl (0) or Wr (1). N_NODES = 3125*16 exactly -> EXEC all-ones.
// ---------------------------------------------------------------------------
__global__ __launch_bounds__(256)
void node_xform_kernel(const float* __restrict__ x,
                       const float* __restrict__ Wl,
                       const float* __restrict__ Wr,
                       float* __restrict__ xl,
                       float* __restrict__ xr) {
    __shared__ float sW[F_IN * WSTRIDE];               // 36 KB, padded rows

    const float* __restrict__ W   = (blockIdx.y == 0) ? Wl : Wr;
    float*       __restrict__ out = (blockIdx.y == 0) ? xl : xr;

    // ---- stage W[128][64] -> sW[128][72] (all 256 threads, 32 elems each) ----
#ifdef HAVE_ASYNC_LDS
    for (int i = threadIdx.x; i < F_IN * HID; i += 256) {
        const int r = i >> 6, c = i & 63;
        __builtin_amdgcn_global_load_async_to_lds_b32(
            AS_GLOBAL_I(W) + i, AS_LDS_I(sW) + r * WSTRIDE + c, 0, 0);
    }
#ifdef HAVE_WAIT_ASYNC
    __builtin_amdgcn_s_wait_asynccnt(0);
#else
    asm volatile("s_wait_asynccnt 0x0" ::: "memory");
#endif
#else
    for (int i = threadIdx.x; i < F_IN * HID; i += 256) {
        const int r = i >> 6, c = i & 63;
        sW[r * WSTRIDE + c] = W[i];
    }
#endif
    __syncthreads();

    const int wave = threadIdx.x >> 5;
    const int lane = threadIdx.x & 31;
    const int tile = blockIdx.x * 8 + wave;            // 16-row tile index
    if (tile * 16 >= N_NODES) return;                  // wave-uniform exit

    const int half = lane >> 4;        // 0: lanes 0-15, 1: lanes 16-31
    const int l15  = lane & 15;
    const int arow = tile * 16 + l15;  // A-matrix row for this lane
    const int kk   = half * 2;         // K sub-offset (0 or 2)

    v8f acc[4] = {};

    for (int k0 = 0; k0 < F_IN; k0 += 4) {
        v2f a;
        a.x = x[arow * F_IN + k0 + kk];
        a.y = x[arow * F_IN + k0 + kk + 1];
#pragma unroll
        for (int c = 0; c < 4; ++c) {
            const int col = c * 16 + l15;
            v2f b;
            b.x = sW[(k0 + kk)     * WSTRIDE + col];
            b.y = sW[(k0 + kk + 1) * WSTRIDE + col];
            acc[c] = __builtin_amdgcn_wmma_f32_16x16x4_f32(
                false, a, false, b, (short)0, acc[c], false, false);
        }
    }

#pragma unroll
    for (int c = 0; c < 4; ++c) {
        const int col = c * 16 + l15;
#pragma unroll
        for (int j = 0; j < 8; ++j) {
            const int row = tile * 16 + half * 8 + j;  // C/D: VGPR j -> M=j / M=8+j
            out[row * HID + col] = acc[c][j];
        }
    }
}

// ---------------------------------------------------------------------------
// a = leaky_relu(image @ W1 + b1, 0.01) -> d_out[:, 0:64]   (WMMA f32 16x16x4)
// W1 (230 KB) too large to stage; G = 32*16 exactly; K = 900 = 225*4.
// ---------------------------------------------------------------------------
__global__ __launch_bounds__(256)
void image_linear_kernel(const float* __restrict__ image,
                         const float* __restrict__ W1,
                         const float* __restrict__ b1,
                         float* __restrict__ out) {
    const int wave = threadIdx.x >> 5;
    const int lane = threadIdx.x & 31;
    const int tile = blockIdx.x * 8 + wave;
    if (tile * 16 >= N_GRAPH) return;

    const int half = lane >> 4;
    const int l15  = lane & 15;
    const int arow = tile * 16 + l15;
    const int kk   = half * 2;

    v8f acc[4] = {};

    for (int k0 = 0; k0 < 900; k0 += 4) {
        v2f a;
        a.x = image[arow * 900 + k0 + kk];
        a.y = image[arow * 900 + k0 + kk + 1];
#pragma unroll
        for (int c = 0; c < 4; ++c) {
            const int col = c * 16 + l15;
            v2f b;
            b.x = W1[(k0 + kk)     * HID + col];
            b.y = W1[(k0 + kk + 1) * HID + col];
            acc[c] = __builtin_amdgcn_wmma_f32_16x16x4_f32(
                false, a, false, b, (short)0, acc[c], false, false);
        }
    }

#pragma unroll
    for (int c = 0; c < 4; ++c) {
        const int col = c * 16 + l15;
        const float bb = b1[col];
#pragma unroll
        for (int j = 0; j < 8; ++j) {
            const int row = tile * 16 + half * 8 + j;
            float v = acc[c][j] + bb;
            v = (v > 0.0f) ? v : 0.01f * v;
            out[row * (2 * HID) + col] = v;   // columns 0..63 of concat output
        }
    }
}

// ---------------------------------------------------------------------------
// Pass A: per-edge attention logit + segment max (ordered-uint atomicMax).
// One wave32 per edge; lane handles h = lane and h = lane+32.
// ---------------------------------------------------------------------------
__global__ __launch_bounds__(256)
void edge_logit_kernel(const float* __restrict__ xl,
                       const float* __restrict__ xr,
                       const int*   __restrict__ ei,
                       const float* __restrict__ ea,
                       const float* __restrict__ We,
                       const float* __restrict__ att,
                       float* __restrict__ logit,
                       unsigned int* __restrict__ mkey) {
    __shared__ float sWe[EDIM * HID];
    __shared__ float sAtt[HID];
    for (int i = threadIdx.x; i < EDIM * HID; i += blockDim.x) sWe[i] = We[i];
    if (threadIdx.x < HID) sAtt[threadIdx.x] = att[threadIdx.x];
    __syncthreads();

    const int e = blockIdx.x * 8 + (threadIdx.x >> 5);
    if (e >= N_EDGES) return;
    const int lane = threadIdx.x & 31;
    const int src = ei[e];
    const int dst = ei[N_EDGES + e];

    float eav[EDIM];
#pragma unroll
    for (int d = 0; d < EDIM; ++d) eav[d] = ea[e * EDIM + d];

    float s = 0.0f;
#pragma unroll
    for (int t = 0; t < 2; ++t) {
        const int h = lane + t * 32;
        float v = xl[src * HID + h] + xr[dst * HID + h];
#pragma unroll
        for (int d = 0; d < EDIM; ++d) v += eav[d] * sWe[d * HID + h];
        v = (v > 0.0f) ? v : 0.2f * v;            // leaky_relu 0.2
        s += v * sAtt[h];
    }
#pragma unroll
    for (int off = 16; off > 0; off >>= 1)        // wave32 reduction
        s += __shfl_xor(s, off, 32);

    if (lane == 0) {
        logit[e] = s;
        unsigned int b   = __float_as_uint(s);
        unsigned int key = (b & 0x80000000u) ? ~b : (b | 0x80000000u);
        atomicMax(&mkey[dst], key);
    }
}

// ---------------------------------------------------------------------------
// Pass B: p = exp(logit - m[dst]); denom[dst] += p
// ---------------------------------------------------------------------------
__global__ void edge_softmax_kernel(const float* __restrict__ logit,
                                    const int*   __restrict__ ei,
                                    const unsigned int* __restrict__ mkey,
                                    float* __restrict__ p,
                                    float* __restrict__ denom) {
    const int e = blockIdx.x * blockDim.x + threadIdx.x;
    if (e >= N_EDGES) return;
    const int dst = ei[N_EDGES + e];
    const unsigned int k = mkey[dst];
    const unsigned int b = (k & 0x80000000u) ? (k & 0x7fffffffu) : ~k;
    const float m  = __uint_as_float(b);
    const float pe = __expf(logit[e] - m);
    p[e] = pe;
    atomicAdd(&denom[dst], pe);
}

// ---------------------------------------------------------------------------
// Pass C: nodeout[dst] += (p/denom) * xl[src]   (one thread per edge-channel)
// ---------------------------------------------------------------------------
__global__ void edge_aggregate_kernel(const float* __restrict__ xl,
                                      const int*   __restrict__ ei,
                                      const float* __restrict__ p,
                                      const float* __restrict__ denom,
                                      float* __restrict__ nodeout) {
    const long long idx = (long long)blockIdx.x * blockDim.x + threadIdx.x;
    if (idx >= (long long)N_EDGES * HID) return;
    const int e = (int)(idx >> 6);
    const int h = (int)(idx & 63);
    const int src = ei[e];
    const int dst = ei[N_EDGES + e];
    const float alpha = p[e] / denom[dst];
    atomicAdd(&nodeout[dst * HID + h], alpha * xl[src * HID + h]);
}

// ---------------------------------------------------------------------------
// Pass D: h = relu(nodeout + bias); scatter-sum into per-graph pools + counts
// ---------------------------------------------------------------------------
__global__ void node_pool_kernel(const float* __restrict__ nodeout,
                                 const float* __restrict__ bias,
                                 const int*   __restrict__ batch,
                                 float* __restrict__ sums,
                                 float* __restrict__ cnts) {
    const int idx = blockIdx.x * blockDim.x + threadIdx.x;
    if (idx >= N_NODES * HID) return;
    const int n = idx >> 6;
    const int h = idx & 63;
    float v = nodeout[idx] + bias[h];
    v = (v > 0.0f) ? v : 0.0f;
    const int g = batch[n];
    atomicAdd(&sums[g * HID + h], v);
    if (h == 0) atomicAdd(&cnts[g], 1.0f);
}

// ---------------------------------------------------------------------------
// Pass E: pooled = sums / max(cnts, 1) -> d_out[:, 64:128]
// ---------------------------------------------------------------------------
__global__ void pool_finalize_kernel(const float* __restrict__ sums,
                                     const float* __restrict__ cnts,
                                     float* __restrict__ out) {
    const int idx = blockIdx.x * blockDim.x + threadIdx.x;
    if (idx >= N_GRAPH * HID) return;
    const int g = idx >> 6;
    const int h = idx & 63;
    float c = cnts[g];
    c = (c > 1.0f) ? c : 1.0f;
    out[g * (2 * HID) + HID + h] = sums[idx] / c;
}

// ---------------------------------------------------------------------------
extern "C" void kernel_launch(void* const* d_in, const int* in_sizes, int n_in,
                              void* d_out, int out_size, void* d_ws, size_t ws_size,
                              hipStream_t stream) {
    const float* x     = (const float*)d_in[0];
    const int*   ei    = (const int*)  d_in[1];
    const float* ea    = (const float*)d_in[2];
    const float* image = (const float*)d_in[3];
    const int*   batch = (const int*)  d_in[4];
    const float* Wl    = (const float*)d_in[5];
    const float* Wr    = (const float*)d_in[6];
    const float* We    = (const float*)d_in[7];
    const float* att   = (const float*)d_in[8];
    const float* bias  = (const float*)d_in[9];
    const float* W1    = (const float*)d_in[10];
    const float* b1    = (const float*)d_in[11];
    float* out = (float*)d_out;

    // ---- workspace carve-out ----
    float* ws = (float*)d_ws;
    float* xl      = ws;  ws += (size_t)N_NODES * HID;
    float* xr      = ws;  ws += (size_t)N_NODES * HID;
    float* nodeout = ws;  ws += (size_t)N_NODES * HID;   // zeroed region starts here
    float* logit   = ws;  ws += N_EDGES;
    float* p       = ws;  ws += N_EDGES;
    float* denom   = ws;  ws += N_NODES;
    unsigned int* mkey = (unsigned int*)ws; ws += N_NODES;
    float* sums    = ws;  ws += N_GRAPH * HID;
    float* cnts    = ws;  ws += N_GRAPH;

    // zero all accumulators (nodeout..cnts contiguous); mkey==0 is the min ordered key
    const size_t zero_elems = (size_t)N_NODES * HID + 2ULL * N_EDGES +
                              2ULL * N_NODES + (size_t)N_GRAPH * HID + N_GRAPH;
    (void)hipMemsetAsync(nodeout, 0, zero_elems * sizeof(float), stream);

    // ---- dense transforms (WMMA f32) ----
    {
        dim3 grid((3125 + 7) / 8, 2);                 // 3125 row tiles, y: Wl/Wr
        node_xform_kernel<<<grid, 256, 0, stream>>>(x, Wl, Wr, xl, xr);
    }
    image_linear_kernel<<<4, 256, 0, stream>>>(image, W1, b1, out);  // 32 tiles / 8 waves

    // ---- GATv2 edge pipeline ----
    edge_logit_kernel<<<(N_EDGES + 7) / 8, 256, 0, stream>>>(
        xl, xr, ei, ea, We, att, logit, mkey);
    edge_softmax_kernel<<<(N_EDGES + 255) / 256, 256, 0, stream>>>(
        logit, ei, mkey, p, denom);
    {
        long long tot = (long long)N_EDGES * HID;
        edge_aggregate_kernel<<<(unsigned)((tot + 255) / 256), 256, 0, stream>>>(
            xl, ei, p, denom, nodeout);
    }

    // ---- relu + mean pool ----
    node_pool_kernel<<<(N_NODES * HID + 255) / 256, 256, 0, stream>>>(
        nodeout, bias, batch, sums, cnts);
    pool_finalize_kernel<<<(N_GRAPH * HID + 255) / 256, 256, 0, stream>>>(
        sums, cnts, out);
}